// GAT_56873956933641
// MI455X (gfx1250) — compile-verified
//
#include <hip/hip_runtime.h>
#include <math.h>

#define N_NODES  20000
#define N_EDGES  320000
#define N_GRAPHS 128
#define DD 64
#define HH 8
#define FF 64
#define HF 512   // H*F

typedef float v2f __attribute__((ext_vector_type(2)));
typedef float v8f __attribute__((ext_vector_type(8)));

// ---------------------------------------------------------------- utilities
__device__ __forceinline__ void atomicMaxF(float* addr, float val) {
    int* ai = (int*)addr;
    int old = __float_as_int(*addr);
    while (true) {
        float fold = __int_as_float(old);
        if (fold >= val) break;                 // NaN poison also gets replaced (false)
        int assumed = old;
        old = atomicCAS(ai, assumed, __float_as_int(val));
        if (old == assumed) break;
    }
}

__global__ void k_fill(float* p, long n, float v) {
    long t = (long)blockIdx.x * blockDim.x + threadIdx.x;
    if (t < n) p[t] = v;
}

// ---------------------------------------------------------------- degrees / norms
__global__ void k_degrees(const int* __restrict__ src, const int* __restrict__ dst,
                          float* odeg, float* ideg) {
    int e = blockIdx.x * blockDim.x + threadIdx.x;
    if (e >= N_EDGES) return;
    atomicAdd(&odeg[src[e]], 1.0f);
    atomicAdd(&ideg[dst[e]], 1.0f);
}

__global__ void k_norms(float* ns, float* nd) {
    int i = blockIdx.x * blockDim.x + threadIdx.x;
    if (i >= N_NODES) return;
    ns[i] = rsqrtf(fmaxf(ns[i], 1.0f));
    nd[i] = rsqrtf(fmaxf(nd[i], 1.0f));
}

// ---------------------------------------------------------------- GraphConv scatter
__global__ void k_gc_scatter(const float* __restrict__ x, const int* __restrict__ src,
                             const int* __restrict__ dst, const float* __restrict__ ns,
                             float* agg) {
    long t = (long)blockIdx.x * blockDim.x + threadIdx.x;
    if (t >= (long)N_EDGES * DD) return;
    int e = (int)(t >> 6), f = (int)(t & 63);
    int s = src[e], d = dst[e];
    atomicAdd(&agg[(long)d * DD + f], x[(long)s * DD + f] * ns[s]);
}

// ---------------------------------------------------------------- WMMA GEMM
// C[M,Nc] = act( diag(rowscale) * A[M,K] @ B + bias )
//   K      : compile-time depth (multiple of 4) -> no remainder/zero-trip code
//   BTRANS : B stored [Nc,K], used transposed (else row-major [K,Nc])
//   ACT    : 1 = relu
//   HASRS  : multiply A rows by rowscale
//   HASBIAS: add bias[col]
//   LDSB   : stage whole B in LDS (requires B element count <= 4096)
// One wave per 16x16 C tile, V_WMMA_F32_16X16X4_F32, K stepped by 4.
// M, Nc multiples of 16.  All early-exits are wave-uniform (EXEC all-ones at WMMA).
template<int K, int BTRANS, int ACT, int HASRS, int HASBIAS, int LDSB>
__global__ void k_wmma_gemm(const float* __restrict__ A, const float* __restrict__ Bm,
                            const float* __restrict__ bias, const float* __restrict__ rowscale,
                            float* __restrict__ C, int M, int Nc) {
    __shared__ float sB[LDSB ? 4096 : 1];
    if (LDSB) {
        int total = BTRANS ? Nc * K : K * Nc;
        for (int i = threadIdx.x; i < total; i += blockDim.x) sB[i] = Bm[i];
        __syncthreads();
    }

    int wavesPerBlock = blockDim.x >> 5;
    int wave = blockIdx.x * wavesPerBlock + (threadIdx.x >> 5);
    int lane = threadIdx.x & 31;
    int tilesN = Nc >> 4;
    int tilesM = M >> 4;
    if (wave >= tilesM * tilesN) return;        // wave-uniform: EXEC stays all-ones
    int tm = wave / tilesN, tn = wave % tilesN;

    int half2 = (lane >> 4) << 1;               // 0: K pair {0,1}, 1: K pair {2,3}
    int l15   = lane & 15;
    int rowA  = tm * 16 + l15;                  // A: M striped over lanes 0..15
    int colB  = tn * 16 + l15;                  // B: N striped over lanes 0..15
    float rs  = HASRS ? rowscale[rowA] : 1.0f;

    const float* Arow = A + (long)rowA * K;

    v8f acc = {0.f, 0.f, 0.f, 0.f, 0.f, 0.f, 0.f, 0.f};
#pragma unroll 4
    for (int k0 = 0; k0 < K; k0 += 4) {
        int ka = k0 + half2;
        v2f a = *(const v2f*)(Arow + ka);        // 8B-aligned global_load_b64
        if (HASRS) { a.x *= rs; a.y *= rs; }
        v2f b;
        if (LDSB) {
            if (BTRANS) { b.x = sB[colB * K + ka];  b.y = sB[colB * K + ka + 1]; }
            else        { b.x = sB[ka * Nc + colB]; b.y = sB[(ka + 1) * Nc + colB]; }
        } else {
            if (BTRANS) { b = *(const v2f*)(Bm + (long)colB * K + ka); }
            else        { b.x = Bm[(long)ka * Nc + colB];
                          b.y = Bm[(long)(ka + 1) * Nc + colB]; }
        }
        acc = __builtin_amdgcn_wmma_f32_16x16x4_f32(
            false, a, false, b, (short)0, acc, false, false);
    }
    float bv = HASBIAS ? bias[colB] : 0.0f;
#pragma unroll
    for (int v = 0; v < 8; ++v) {
        int row = tm * 16 + v + (half2 << 2);   // C/D: VGPR v -> M=v (lanes<16) / v+8
        float val = acc[v] + bv;
        if (ACT == 1) val = fmaxf(val, 0.0f);
        C[(long)row * Nc + colB] = val;
    }
}

// ---------------------------------------------------------------- GAT attention
__global__ void k_el_er(const float* __restrict__ feat, const float* __restrict__ al,
                        const float* __restrict__ ar, float* el, float* er) {
    int t = blockIdx.x * blockDim.x + threadIdx.x;
    if (t >= N_NODES * HH) return;
    int n = t >> 3, h = t & 7;
    const float* fp  = feat + (long)n * HF + h * FF;
    const float* alp = al + h * FF;
    const float* arp = ar + h * FF;
    float sl = 0.f, sr = 0.f;
    for (int f = 0; f < FF; ++f) { float v = fp[f]; sl += v * alp[f]; sr += v * arp[f]; }
    el[t] = sl; er[t] = sr;
}

__global__ void k_score_max(const int* __restrict__ src, const int* __restrict__ dst,
                            const float* __restrict__ el, const float* __restrict__ er,
                            float* score, float* smax) {
    int t = blockIdx.x * blockDim.x + threadIdx.x;
    if (t >= N_EDGES * HH) return;
    int e = t >> 3, h = t & 7;
    float s = el[src[e] * HH + h] + er[dst[e] * HH + h];
    s = (s > 0.f) ? s : 0.2f * s;               // leaky_relu(0.2)
    score[t] = s;
    atomicMaxF(&smax[dst[e] * HH + h], s);
}

__global__ void k_fix_neginf(float* p, long n) {
    long t = (long)blockIdx.x * blockDim.x + threadIdx.x;
    if (t < n && p[t] == -INFINITY) p[t] = 0.f;
}

__global__ void k_exp_sum(const int* __restrict__ dst, const float* __restrict__ smax,
                          float* score, float* ssum) {
    int t = blockIdx.x * blockDim.x + threadIdx.x;
    if (t >= N_EDGES * HH) return;
    int e = t >> 3, h = t & 7;
    float ex = __expf(score[t] - smax[dst[e] * HH + h]);
    score[t] = ex;                              // score buffer now holds ex
    atomicAdd(&ssum[dst[e] * HH + h], ex);
}

__global__ void k_attn_scatter(const int* __restrict__ src, const int* __restrict__ dst,
                               const float* __restrict__ ex, const float* __restrict__ ssum,
                               const float* __restrict__ feat, float* rst) {
    long t = (long)blockIdx.x * blockDim.x + threadIdx.x;
    if (t >= (long)N_EDGES * HF) return;
    int e = (int)(t >> 9), c = (int)(t & 511), h = c >> 6;
    int d = dst[e], s = src[e];
    float alpha = ex[(long)e * HH + h] / (ssum[d * HH + h] + 1e-16f);
    atomicAdd(&rst[(long)d * HF + c], feat[(long)s * HF + c] * alpha);
}

__global__ void k_elu_bias(float* rst, const float* __restrict__ gat_b) {
    long t = (long)blockIdx.x * blockDim.x + threadIdx.x;
    if (t >= (long)N_NODES * HF) return;
    int c = (int)(t & 511);
    float v = rst[t] + gat_b[c];
    rst[t] = (v > 0.f) ? v : expm1f(v);         // elu, in place: rst becomes h
}

// ---------------------------------------------------------------- pooling
__global__ void k_gate(const float* __restrict__ h, const float* __restrict__ gw,
                       const float* __restrict__ gb, float* gate) {
    int n = blockIdx.x * blockDim.x + threadIdx.x;
    if (n >= N_NODES) return;
    float s = gb[0];
    const float* hp = h + (long)n * HF;
    for (int c = 0; c < HF; ++c) s += hp[c] * gw[c];
    gate[n] = s;
}

__global__ void k_gate_max(const int* __restrict__ gid, const float* __restrict__ gate,
                           float* gmax) {
    int n = blockIdx.x * blockDim.x + threadIdx.x;
    if (n >= N_NODES) return;
    atomicMaxF(&gmax[gid[n]], gate[n]);
}

__global__ void k_gate_exp(const int* __restrict__ gid, const float* __restrict__ gate,
                           const float* __restrict__ gmax, float* ge, float* gs) {
    int n = blockIdx.x * blockDim.x + threadIdx.x;
    if (n >= N_NODES) return;
    float e = __expf(gate[n] - gmax[gid[n]]);
    ge[n] = e;
    atomicAdd(&gs[gid[n]], e);
}

__global__ void k_pool(const int* __restrict__ gid, const float* __restrict__ h,
                       const float* __restrict__ ge, const float* __restrict__ gs,
                       float* h1p, float* h2p) {
    long t = (long)blockIdx.x * blockDim.x + threadIdx.x;
    if (t >= (long)N_NODES * HF) return;
    int n = (int)(t >> 9), c = (int)(t & 511);
    int g = gid[n];
    float a = ge[n] / (gs[g] + 1e-16f);
    float v = h[t];
    atomicAdd(&h1p[(long)g * HF + c], v * a);
    atomicMaxF(&h2p[(long)g * HF + c], v);
}

// ---------------------------------------------------------------- head
__global__ void k_pfix_bn(const float* __restrict__ h1p, const float* __restrict__ h2p,
                          const float* __restrict__ bn_g, const float* __restrict__ bn_b,
                          const float* __restrict__ bn_m, const float* __restrict__ bn_v,
                          float* out_p, float* xbn) {
    int t = blockIdx.x * blockDim.x + threadIdx.x;
    if (t >= N_GRAPHS * 1024) return;
    int b = t >> 10, c = t & 1023;
    float v = (c < HF) ? h1p[b * HF + c] : h2p[b * HF + (c - HF)];
    if (c >= HF && v == -INFINITY) v = 0.f;     // max over empty segment
    float p = (v > 0.f) ? v : expm1f(v);        // elu
    out_p[t] = p;
    xbn[t] = (p - bn_m[c]) * rsqrtf(bn_v[c] + 1e-5f) * bn_g[c] + bn_b[c];
}

__global__ void k_fc3(const float* __restrict__ x2, const float* __restrict__ w,
                      const float* __restrict__ b, float* out) {
    int bb = blockIdx.x * blockDim.x + threadIdx.x;
    if (bb >= N_GRAPHS) return;
    float s = b[0];
    for (int k = 0; k < 32; ++k) s += x2[bb * 32 + k] * w[k];
    out[bb] = s;
}

// ---------------------------------------------------------------- launch
static inline unsigned cdivu(long a, int b) { return (unsigned)((a + b - 1) / b); }

extern "C" void kernel_launch(void* const* d_in, const int* in_sizes, int n_in,
                              void* d_out, int out_size, void* d_ws, size_t ws_size,
                              hipStream_t stream) {
    const float* xn     = (const float*)d_in[0];
    const int*   eidx   = (const int*)  d_in[2];
    const int*   gid    = (const int*)  d_in[3];
    const float* W1     = (const float*)d_in[4];
    const float* b1     = (const float*)d_in[5];
    const float* W2     = (const float*)d_in[6];
    const float* b2     = (const float*)d_in[7];
    const float* gatW   = (const float*)d_in[8];
    const float* attn_l = (const float*)d_in[9];
    const float* attn_r = (const float*)d_in[10];
    const float* gat_b  = (const float*)d_in[11];
    const float* gateW  = (const float*)d_in[12];
    const float* gateb  = (const float*)d_in[13];
    const float* bn_g   = (const float*)d_in[14];
    const float* bn_b   = (const float*)d_in[15];
    const float* bn_m   = (const float*)d_in[16];
    const float* bn_v   = (const float*)d_in[17];
    const float* fc1W   = (const float*)d_in[18];
    const float* fc1b   = (const float*)d_in[19];
    const float* fc2W   = (const float*)d_in[20];
    const float* fc2b   = (const float*)d_in[21];
    const float* fc3W   = (const float*)d_in[22];
    const float* fc3b   = (const float*)d_in[23];

    const int* src = eidx;              // edge_index[0]
    const int* dst = eidx + N_EDGES;    // edge_index[1]

    float* out   = (float*)d_out;       // [0..128): h_out, [128..): p (B x 1024)
    float* out_p = out + N_GRAPHS;

    // workspace carve-up (floats)
    float* ws     = (float*)d_ws;
    float* norm_s = ws;                               // N
    float* norm_d = norm_s + N_NODES;                 // N
    float* agg    = norm_d + N_NODES;                 // N*64
    float* hA     = agg    + (long)N_NODES * DD;      // N*64
    float* feat   = hA     + (long)N_NODES * DD;      // N*512
    float* el     = feat   + (long)N_NODES * HF;      // N*8
    float* er     = el     + (long)N_NODES * HH;      // N*8
    float* smax   = er     + (long)N_NODES * HH;      // N*8
    float* ssum   = smax   + (long)N_NODES * HH;      // N*8
    float* score  = ssum   + (long)N_NODES * HH;      // E*8  (score -> ex)
    float* rst    = score  + (long)N_EDGES * HH;      // N*512 (rst -> h)
    float* gate   = rst    + (long)N_NODES * HF;      // N
    float* ge     = gate   + N_NODES;                 // N
    float* gmax   = ge     + N_NODES;                 // B
    float* gs     = gmax   + N_GRAPHS;                // B
    float* h1p    = gs     + N_GRAPHS;                // B*512
    float* h2p    = h1p    + (long)N_GRAPHS * HF;     // B*512
    float* xbn    = h2p    + (long)N_GRAPHS * HF;     // B*1024
    float* x1     = xbn    + (long)N_GRAPHS * 1024;   // B*64
    float* x2     = x1     + (long)N_GRAPHS * 64;     // B*32

    const int T = 256;

    // degrees -> norm_s / norm_d
    hipMemsetAsync(norm_s, 0, sizeof(float) * 2 * N_NODES, stream);
    k_degrees<<<cdivu(N_EDGES, T), T, 0, stream>>>(src, dst, norm_s, norm_d);
    k_norms<<<cdivu(N_NODES, T), T, 0, stream>>>(norm_s, norm_d);

    // GraphConv 1: scatter, then relu(diag(norm_d)*agg @ W1 + b1)  (B staged in LDS)
    hipMemsetAsync(agg, 0, sizeof(float) * (long)N_NODES * DD, stream);
    k_gc_scatter<<<cdivu((long)N_EDGES * DD, T), T, 0, stream>>>(xn, src, dst, norm_s, agg);
    {
        int waves = (N_NODES / 16) * (DD / 16);
        k_wmma_gemm<DD, 0, 1, 1, 1, 1><<<cdivu(waves, 8), 256, 0, stream>>>(
            agg, W1, b1, norm_d, hA, N_NODES, DD);
    }
    // GraphConv 2
    hipMemsetAsync(agg, 0, sizeof(float) * (long)N_NODES * DD, stream);
    k_gc_scatter<<<cdivu((long)N_EDGES * DD, T), T, 0, stream>>>(hA, src, dst, norm_s, agg);
    {
        int waves = (N_NODES / 16) * (DD / 16);
        k_wmma_gemm<DD, 0, 1, 1, 1, 1><<<cdivu(waves, 8), 256, 0, stream>>>(
            agg, W2, b2, norm_d, hA, N_NODES, DD);
    }

    // GAT projection: feat = hA @ gatW^T   (gatW stored [512,64]; too big for LDS stage)
    {
        int waves = (N_NODES / 16) * (HF / 16);
        k_wmma_gemm<DD, 1, 0, 0, 0, 0><<<cdivu(waves, 8), 256, 0, stream>>>(
            hA, gatW, nullptr, nullptr, feat, N_NODES, HF);
    }
    k_el_er<<<cdivu(N_NODES * HH, T), T, 0, stream>>>(feat, attn_l, attn_r, el, er);

    // segment softmax over incoming edges
    k_fill<<<cdivu((long)N_NODES * HH, T), T, 0, stream>>>(smax, (long)N_NODES * HH, -INFINITY);
    hipMemsetAsync(ssum, 0, sizeof(float) * (long)N_NODES * HH, stream);
    k_score_max<<<cdivu(N_EDGES * HH, T), T, 0, stream>>>(src, dst, el, er, score, smax);
    k_fix_neginf<<<cdivu((long)N_NODES * HH, T), T, 0, stream>>>(smax, (long)N_NODES * HH);
    k_exp_sum<<<cdivu(N_EDGES * HH, T), T, 0, stream>>>(dst, smax, score, ssum);

    // weighted aggregation + bias + elu
    hipMemsetAsync(rst, 0, sizeof(float) * (long)N_NODES * HF, stream);
    k_attn_scatter<<<cdivu((long)N_EDGES * HF, T), T, 0, stream>>>(src, dst, score, ssum,
                                                                   feat, rst);
    k_elu_bias<<<cdivu((long)N_NODES * HF, T), T, 0, stream>>>(rst, gat_b);

    // global attention pooling + max pooling
    k_gate<<<cdivu(N_NODES, T), T, 0, stream>>>(rst, gateW, gateb, gate);
    k_fill<<<cdivu(N_GRAPHS, T), T, 0, stream>>>(gmax, N_GRAPHS, -INFINITY);
    hipMemsetAsync(gs, 0, sizeof(float) * N_GRAPHS, stream);
    k_gate_max<<<cdivu(N_NODES, T), T, 0, stream>>>(gid, gate, gmax);
    k_fix_neginf<<<cdivu(N_GRAPHS, T), T, 0, stream>>>(gmax, N_GRAPHS);
    k_gate_exp<<<cdivu(N_NODES, T), T, 0, stream>>>(gid, gate, gmax, ge, gs);

    hipMemsetAsync(h1p, 0, sizeof(float) * (long)N_GRAPHS * HF, stream);
    k_fill<<<cdivu((long)N_GRAPHS * HF, T), T, 0, stream>>>(h2p, (long)N_GRAPHS * HF, -INFINITY);
    k_pool<<<cdivu((long)N_NODES * HF, T), T, 0, stream>>>(gid, rst, ge, gs, h1p, h2p);

    // head: elu(concat) -> p (output), batchnorm -> xbn
    k_pfix_bn<<<cdivu(N_GRAPHS * 1024, T), T, 0, stream>>>(h1p, h2p, bn_g, bn_b, bn_m, bn_v,
                                                           out_p, xbn);
    // fc1: [128,1024] @ fc1W^T -> [128,64], relu  (WMMA, B transposed, B too big for LDS)
    {
        int waves = (N_GRAPHS / 16) * (64 / 16);
        k_wmma_gemm<1024, 1, 1, 0, 1, 0><<<cdivu(waves, 8), 256, 0, stream>>>(
            xbn, fc1W, fc1b, nullptr, x1, N_GRAPHS, 64);
    }
    // fc2: [128,64] @ fc2W^T -> [128,32], relu  (WMMA, B staged in LDS)
    {
        int waves = (N_GRAPHS / 16) * (32 / 16);
        k_wmma_gemm<DD, 1, 1, 0, 1, 1><<<cdivu(waves, 8), 256, 0, stream>>>(
            x1, fc2W, fc2b, nullptr, x2, N_GRAPHS, 32);
    }
    // fc3: [128,32] @ [32,1] -> h_out
    k_fc3<<<cdivu(N_GRAPHS, T), T, 0, stream>>>(x2, fc3W, fc3b, out);
}